// LocalSpatialEncoding_87943750352950
// MI455X (gfx1250) — compile-verified
//
#include <hip/hip_runtime.h>
#include <hip/hip_bf16.h>

// ---------------------------------------------------------------------------
// LocalSpatialEncoding on MI455X (gfx1250, wave32).
//   K1 pack      : xyz -> float4 (x,y,z,|p|^2)
//   K2 knn       : brute-force top-16; candidate range split 4-way per query
//                  (4x wave occupancy), partial lists merged in LDS
//   K3 stats1    : feat -> z1 via v_wmma_f32_16x16x4_f32, per-channel sum/sumsq
//   K3b reduce   : deterministic fixed-order reduction of block partials
//   K4 stats2    : feat -> z1 -> bn1+relu -> z2 (WMMA), stats of z2
//   K4b reduce
//   K5 final     : full recompute, bn2+relu, transposed store (b,d,n,k)
// ---------------------------------------------------------------------------

#define NPTS   8192
#define BATCH  2
#define KNNK   16
#define DOUT   64
#define ROWS   (BATCH * NPTS * KNNK)     // 262144
#define TILES  (ROWS / 16)               // 16384 == BATCH*NPTS (one tile = one query's 16 neighbors)
#define WAVES  4                         // waves per block (wave32 -> 128 threads)
#define TPW    4                         // tiles per wave
#define NBLK   (TILES / (WAVES * TPW))   // 1024 blocks for the WMMA kernels
#define FS     13                        // feat LDS row stride (10 ch padded to 12, +1 anti-conflict)
#define HS     68                        // hidden LDS row stride (64 + 4)
#define W2S    65                        // W2 LDS row stride
#define INV_CNT (1.0f / 262144.0f)
#define BN_EPS  1e-5f

// KNN split configuration
#define QSPLIT 4                          // threads per query
#define QPB    64                         // queries per 256-thread block
#define CHUNK  (NPTS / QSPLIT)            // 2048 candidates per thread
#define CTILE  64                         // candidates staged per split per step

typedef float v2f __attribute__((ext_vector_type(2)));
typedef float v8f __attribute__((ext_vector_type(8)));

__device__ __forceinline__ v8f wmma4(v2f a, v2f b, v8f c) {
  // D = A(16x4 f32) * B(4x16 f32) + C(16x16 f32)
  return __builtin_amdgcn_wmma_f32_16x16x4_f32(false, a, false, b, (short)0, c,
                                               false, false);
}

// ---- spatial feature tile: 16 rows (neighbors of query t) x 12 cols --------
__device__ __forceinline__ void build_feat(float* featw, const float4* xyz4,
                                           const int* knn_idx, int t, int lane) {
  if (lane < 16) {
    float4 ctr = xyz4[t];
    int b = t >> 13;                      // N = 8192
    int nb = knn_idx[t * KNNK + lane];
    float4 nbp = xyz4[b * NPTS + nb];
    float rx = nbp.x - ctr.x, ry = nbp.y - ctr.y, rz = nbp.z - ctr.z;
    float dist = rx * rx + ry * ry + rz * rz;
    float* row = featw + lane * FS;
    row[0] = ctr.x; row[1] = ctr.y; row[2] = ctr.z;
    row[3] = nbp.x; row[4] = nbp.y; row[5] = nbp.z;
    row[6] = rx;    row[7] = ry;    row[8] = rz;
    row[9] = dist;  row[10] = 0.0f; row[11] = 0.0f;
  }
}

// ---- layer 1: 16x12 @ 12x64 -> 4 accumulators of 16x16 ---------------------
__device__ __forceinline__ void layer1(const float* featw, const float* W1lds,
                                       int lane, v8f acc[4]) {
  const v8f zero = {0.f, 0.f, 0.f, 0.f, 0.f, 0.f, 0.f, 0.f};
  int m  = lane & 15;
  int hi = ((lane >> 4) & 1) * 2;        // lanes 16-31 hold K+2 (ISA 16x4 A layout)
#pragma unroll
  for (int nt = 0; nt < 4; ++nt) acc[nt] = zero;
#pragma unroll
  for (int s = 0; s < 3; ++s) {
    int c0 = 4 * s + hi;
    v2f a;
    a.x = featw[m * FS + c0];
    a.y = featw[m * FS + c0 + 1];
#pragma unroll
    for (int nt = 0; nt < 4; ++nt) {
      v2f bf;
      bf.x = W1lds[(nt * 16 + m) * FS + c0];
      bf.y = W1lds[(nt * 16 + m) * FS + c0 + 1];
      acc[nt] = wmma4(a, bf, acc[nt]);
    }
  }
}

// ---- layer 2: 16x64 @ 64x64 ------------------------------------------------
__device__ __forceinline__ void layer2(const float* hldsw, const float* W2lds,
                                       int lane, v8f acc[4]) {
  const v8f zero = {0.f, 0.f, 0.f, 0.f, 0.f, 0.f, 0.f, 0.f};
  int m  = lane & 15;
  int hi = ((lane >> 4) & 1) * 2;
#pragma unroll
  for (int nt = 0; nt < 4; ++nt) acc[nt] = zero;
#pragma unroll
  for (int s = 0; s < 16; ++s) {
    int c0 = 4 * s + hi;
    v2f a;
    a.x = hldsw[m * HS + c0];
    a.y = hldsw[m * HS + c0 + 1];
#pragma unroll
    for (int nt = 0; nt < 4; ++nt) {
      v2f bf;
      bf.x = W2lds[(nt * 16 + m) * W2S + c0];
      bf.y = W2lds[(nt * 16 + m) * W2S + c0 + 1];
      acc[nt] = wmma4(a, bf, acc[nt]);
    }
  }
}

__device__ __forceinline__ void load_w1(float* W1lds, const float* W1, int tid) {
  for (int i = tid; i < 64 * FS; i += blockDim.x) W1lds[i] = 0.0f;
  __syncthreads();
  for (int i = tid; i < 64 * 10; i += blockDim.x)
    W1lds[(i / 10) * FS + (i % 10)] = W1[i];
}

__device__ __forceinline__ void load_w2(float* W2lds, const float* W2, int tid) {
  for (int i = tid; i < 64 * 64; i += blockDim.x)
    W2lds[(i >> 6) * W2S + (i & 63)] = W2[i];
}

__device__ __forceinline__ void bn_consts(const float* st, const float* gamma,
                                          const float* beta, int lane,
                                          float a[4], float b[4]) {
  int cl = lane & 15;
#pragma unroll
  for (int nt = 0; nt < 4; ++nt) {
    int c = nt * 16 + cl;
    float mu  = st[c] * INV_CNT;
    float var = st[64 + c] * INV_CNT - mu * mu;
    float sc  = rsqrtf(var + BN_EPS) * gamma[c];
    a[nt] = sc;
    b[nt] = beta[c] - mu * sc;
  }
}

// z2 accumulators for tile t (feat -> z1 -> bn1+relu -> z2)
__device__ __forceinline__ void tile_z2(int t, int lane, float* featw,
                                        float* hldsw, const float* W1lds,
                                        const float* W2lds, const float4* xyz4,
                                        const int* knn_idx, const float a1[4],
                                        const float b1[4], v8f acc2[4]) {
  build_feat(featw, xyz4, knn_idx, t, lane);
  __syncthreads();
  v8f acc1[4];
  layer1(featw, W1lds, lane, acc1);
  int mo = (lane >= 16) ? 8 : 0;         // C/D layout: vgpr r -> row r (+8 for hi half)
  int cl = lane & 15;
#pragma unroll
  for (int nt = 0; nt < 4; ++nt) {
    int c = nt * 16 + cl;
#pragma unroll
    for (int r = 0; r < 8; ++r) {
      float h = fmaxf(acc1[nt][r] * a1[nt] + b1[nt], 0.0f);
      hldsw[(r + mo) * HS + c] = h;      // D-layout -> A-layout transpose via LDS
    }
  }
  __syncthreads();
  layer2(hldsw, W2lds, lane, acc2);
}

// deterministic per-block stats flush (exactly 2 LDS atomic adds per address)
__device__ __forceinline__ void flush_stats(float* ssum, float* ssq, int wv,
                                            int lane, const float rs[4],
                                            const float rq[4], float* bstats,
                                            int bx, int tid) {
#pragma unroll
  for (int nt = 0; nt < 4; ++nt) {
    int c = nt * 16 + (lane & 15);
    atomicAdd(&ssum[wv * 64 + c], rs[nt]);
    atomicAdd(&ssq[wv * 64 + c], rq[nt]);
  }
  __syncthreads();
  if (tid < 64) {
    float s = 0.f, q = 0.f;
#pragma unroll
    for (int w = 0; w < WAVES; ++w) { s += ssum[w * 64 + tid]; q += ssq[w * 64 + tid]; }
    bstats[bx * 128 + tid]      = s;
    bstats[bx * 128 + 64 + tid] = q;
  }
}

// guarded sorted bubble-insert of (d, idx) into per-thread top-16
__device__ __forceinline__ void topk_insert(float bd[KNNK], int bi[KNNK],
                                            float d, int idx) {
  if (d < bd[KNNK - 1]) {
    float cd = d; int ci = idx;
#pragma unroll
    for (int u = 0; u < KNNK; ++u) {
      if (cd < bd[u]) {
        float td = bd[u]; int ti = bi[u];
        bd[u] = cd; bi[u] = ci; cd = td; ci = ti;
      }
    }
  }
}

// ---------------------------- kernels ---------------------------------------

__global__ void __launch_bounds__(256) k_pack(const float* __restrict__ xyz,
                                              float4* __restrict__ xyz4) {
  int t = blockIdx.x * 256 + threadIdx.x;          // < BATCH*NPTS
  float x = xyz[t * 3 + 0], y = xyz[t * 3 + 1], z = xyz[t * 3 + 2];
  xyz4[t] = make_float4(x, y, z, x * x + y * y + z * z);
}

// 4 threads per query, each scans a 2048-candidate chunk; partials merged in LDS.
__global__ void __launch_bounds__(256) k_knn(const float4* __restrict__ xyz4,
                                             int* __restrict__ knn_idx) {
  __shared__ float4 cand[QSPLIT * CTILE];              //  4 KB
  __shared__ float  mdist[QSPLIT * QPB * KNNK];        // 16 KB
  __shared__ int    midx [QSPLIT * QPB * KNNK];        // 16 KB
  const int tid = threadIdx.x;
  const int ql  = tid & (QPB - 1);                     // query within block
  const int sp  = tid >> 6;                            // split id 0..3
  const int q   = blockIdx.x * QPB + ql;
  const int b   = q >> 13;                             // uniform per block
  float4 p = xyz4[q];
  float bd[KNNK];
  int   bi[KNNK];
#pragma unroll
  for (int j = 0; j < KNNK; ++j) { bd[j] = 3.4e38f; bi[j] = 0; }

  const int cbase = b * NPTS + sp * CHUNK;
  for (int step = 0; step < CHUNK / CTILE; ++step) {   // 32 steps
    cand[tid] = xyz4[cbase + step * CTILE + ql];       // ql doubles as tile slot
    __syncthreads();
#pragma unroll 4
    for (int j = 0; j < CTILE; ++j) {
      float4 cp = cand[sp * CTILE + j];                // wave-uniform -> broadcast
      // same formula as reference: |q|^2 + |c|^2 - 2<q,c>  (tie -> lower index)
      float d = p.w + cp.w - 2.0f * (p.x * cp.x + p.y * cp.y + p.z * cp.z);
      topk_insert(bd, bi, d, sp * CHUNK + step * CTILE + j);
    }
    __syncthreads();
  }

  // publish per-split sorted partial lists
#pragma unroll
  for (int j = 0; j < KNNK; ++j) {
    mdist[(sp * QPB + ql) * KNNK + j] = bd[j];
    midx [(sp * QPB + ql) * KNNK + j] = bi[j];
  }
  __syncthreads();

  // split 0's thread merges: scanning s ascending == global index ascending,
  // strict-< insert preserves jax.top_k tie-break-by-lower-index.
  if (tid < QPB) {
#pragma unroll
    for (int j = 0; j < KNNK; ++j) { bd[j] = 3.4e38f; bi[j] = 0; }
    for (int s = 0; s < QSPLIT; ++s)
      for (int j = 0; j < KNNK; ++j)
        topk_insert(bd, bi, mdist[(s * QPB + ql) * KNNK + j],
                    midx[(s * QPB + ql) * KNNK + j]);
#pragma unroll
    for (int j = 0; j < KNNK; ++j) knn_idx[q * KNNK + j] = bi[j];
  }
}

__global__ void __launch_bounds__(128) k_stats1(const float4* __restrict__ xyz4,
                                                const int* __restrict__ knn_idx,
                                                const float* __restrict__ W1,
                                                float* __restrict__ bstats) {
  __shared__ float W1lds[64 * FS];
  __shared__ float featlds[WAVES][16 * FS];
  __shared__ float ssum[WAVES * 64];
  __shared__ float ssq[WAVES * 64];
  const int tid = threadIdx.x, lane = tid & 31, wv = tid >> 5;
  load_w1(W1lds, W1, tid);
  for (int i = tid; i < WAVES * 64; i += blockDim.x) { ssum[i] = 0.f; ssq[i] = 0.f; }
  __syncthreads();
  float rs[4] = {0, 0, 0, 0}, rq[4] = {0, 0, 0, 0};
  for (int it = 0; it < TPW; ++it) {
    int t = (blockIdx.x * WAVES + wv) * TPW + it;
    __syncthreads();
    build_feat(featlds[wv], xyz4, knn_idx, t, lane);
    __syncthreads();
    v8f acc[4];
    layer1(featlds[wv], W1lds, lane, acc);
#pragma unroll
    for (int nt = 0; nt < 4; ++nt)
#pragma unroll
      for (int r = 0; r < 8; ++r) { float v = acc[nt][r]; rs[nt] += v; rq[nt] += v * v; }
  }
  __syncthreads();
  flush_stats(ssum, ssq, wv, lane, rs, rq, bstats, blockIdx.x, tid);
}

__global__ void __launch_bounds__(128) k_reduce(const float* __restrict__ bstats,
                                                float* __restrict__ stats,
                                                int nblocks) {
  int j = threadIdx.x;                   // 128 threads: [0..63]=sum [64..127]=sumsq
  float s = 0.f;
  for (int b = 0; b < nblocks; ++b) s += bstats[b * 128 + j];   // fixed order
  stats[j] = s;
}

__global__ void __launch_bounds__(128) k_stats2(const float4* __restrict__ xyz4,
                                                const int* __restrict__ knn_idx,
                                                const float* __restrict__ W1,
                                                const float* __restrict__ W2,
                                                const float* __restrict__ st1,
                                                const float* __restrict__ gamma1,
                                                const float* __restrict__ beta1,
                                                float* __restrict__ bstats) {
  __shared__ float W1lds[64 * FS];
  __shared__ float W2lds[64 * W2S];
  __shared__ float featlds[WAVES][16 * FS];
  __shared__ float hlds[WAVES][16 * HS];
  __shared__ float ssum[WAVES * 64];
  __shared__ float ssq[WAVES * 64];
  const int tid = threadIdx.x, lane = tid & 31, wv = tid >> 5;
  load_w1(W1lds, W1, tid);
  load_w2(W2lds, W2, tid);
  for (int i = tid; i < WAVES * 64; i += blockDim.x) { ssum[i] = 0.f; ssq[i] = 0.f; }
  __syncthreads();
  float a1[4], b1[4];
  bn_consts(st1, gamma1, beta1, lane, a1, b1);
  float rs[4] = {0, 0, 0, 0}, rq[4] = {0, 0, 0, 0};
  for (int it = 0; it < TPW; ++it) {
    int t = (blockIdx.x * WAVES + wv) * TPW + it;
    __syncthreads();
    v8f acc2[4];
    tile_z2(t, lane, featlds[wv], hlds[wv], W1lds, W2lds, xyz4, knn_idx, a1, b1, acc2);
#pragma unroll
    for (int nt = 0; nt < 4; ++nt)
#pragma unroll
      for (int r = 0; r < 8; ++r) { float v = acc2[nt][r]; rs[nt] += v; rq[nt] += v * v; }
  }
  __syncthreads();
  flush_stats(ssum, ssq, wv, lane, rs, rq, bstats, blockIdx.x, tid);
}

__global__ void __launch_bounds__(128) k_final(const float4* __restrict__ xyz4,
                                               const int* __restrict__ knn_idx,
                                               const float* __restrict__ W1,
                                               const float* __restrict__ W2,
                                               const float* __restrict__ st1,
                                               const float* __restrict__ gamma1,
                                               const float* __restrict__ beta1,
                                               const float* __restrict__ st2,
                                               const float* __restrict__ gamma2,
                                               const float* __restrict__ beta2,
                                               float* __restrict__ out) {
  __shared__ float W1lds[64 * FS];
  __shared__ float W2lds[64 * W2S];
  __shared__ float featlds[WAVES][16 * FS];
  __shared__ float hlds[WAVES][16 * HS];
  const int tid = threadIdx.x, lane = tid & 31, wv = tid >> 5;
  load_w1(W1lds, W1, tid);
  load_w2(W2lds, W2, tid);
  __syncthreads();
  float a1[4], b1[4], a2[4], b2[4];
  bn_consts(st1, gamma1, beta1, lane, a1, b1);
  bn_consts(st2, gamma2, beta2, lane, a2, b2);
  for (int it = 0; it < TPW; ++it) {
    int t = (blockIdx.x * WAVES + wv) * TPW + it;
    __syncthreads();
    v8f acc2[4];
    tile_z2(t, lane, featlds[wv], hlds[wv], W1lds, W2lds, xyz4, knn_idx, a1, b1, acc2);
    __syncthreads();
    // bn2 + relu, stash transposed tile in LDS
    int mo = (lane >= 16) ? 8 : 0;
    int cl = lane & 15;
#pragma unroll
    for (int nt = 0; nt < 4; ++nt) {
      int c = nt * 16 + cl;
#pragma unroll
      for (int r = 0; r < 8; ++r) {
        float h = fmaxf(acc2[nt][r] * a2[nt] + b2[nt], 0.0f);
        hlds[wv][(r + mo) * HS + c] = h;
      }
    }
    __syncthreads();
    // store out[b][c][n][0..15] as 4x float4 per channel, 2 channels per lane
    int b = t >> 13, n = t & (NPTS - 1);
#pragma unroll
    for (int cc = 0; cc < 2; ++cc) {
      int c = lane + cc * 32;
      float4 v0, v1, v2, v3;
      v0.x = hlds[wv][0 * HS + c];  v0.y = hlds[wv][1 * HS + c];
      v0.z = hlds[wv][2 * HS + c];  v0.w = hlds[wv][3 * HS + c];
      v1.x = hlds[wv][4 * HS + c];  v1.y = hlds[wv][5 * HS + c];
      v1.z = hlds[wv][6 * HS + c];  v1.w = hlds[wv][7 * HS + c];
      v2.x = hlds[wv][8 * HS + c];  v2.y = hlds[wv][9 * HS + c];
      v2.z = hlds[wv][10 * HS + c]; v2.w = hlds[wv][11 * HS + c];
      v3.x = hlds[wv][12 * HS + c]; v3.y = hlds[wv][13 * HS + c];
      v3.z = hlds[wv][14 * HS + c]; v3.w = hlds[wv][15 * HS + c];
      float4* op = (float4*)(out + ((size_t)(b * DOUT + c) * NPTS + n) * KNNK);
      op[0] = v0; op[1] = v1; op[2] = v2; op[3] = v3;
    }
  }
}

// ---------------------------- launch ----------------------------------------

extern "C" void kernel_launch(void* const* d_in, const int* in_sizes, int n_in,
                              void* d_out, int out_size, void* d_ws, size_t ws_size,
                              hipStream_t stream) {
  const float* xyz    = (const float*)d_in[0];
  // d_in[1] = features (unused by the reference)
  const float* W1     = (const float*)d_in[2];
  const float* gamma1 = (const float*)d_in[3];
  const float* beta1  = (const float*)d_in[4];
  const float* W2     = (const float*)d_in[5];
  const float* gamma2 = (const float*)d_in[6];
  const float* beta2  = (const float*)d_in[7];
  float* out = (float*)d_out;

  char* ws = (char*)d_ws;
  float4* xyz4 = (float4*)(ws);                         // 16384 * 16B  = 256 KB
  int*    kidx = (int*)   (ws + 262144);                // 262144 * 4B  = 1 MB
  float*  bst1 = (float*) (ws + 1310720);               // 1024*128*4B  = 512 KB
  float*  st1  = (float*) (ws + 1835008);               // 128 * 4B
  float*  bst2 = (float*) (ws + 1835520);               // 512 KB
  float*  st2  = (float*) (ws + 2359808);               // 128 * 4B      (end ~2.36 MB)

  k_pack  <<<(BATCH * NPTS) / 256, 256, 0, stream>>>(xyz, xyz4);
  k_knn   <<<(BATCH * NPTS) / QPB, 256, 0, stream>>>(xyz4, kidx);
  k_stats1<<<NBLK, 128, 0, stream>>>(xyz4, kidx, W1, bst1);
  k_reduce<<<1, 128, 0, stream>>>(bst1, st1, NBLK);
  k_stats2<<<NBLK, 128, 0, stream>>>(xyz4, kidx, W1, W2, st1, gamma1, beta1, bst2);
  k_reduce<<<1, 128, 0, stream>>>(bst2, st2, NBLK);
  k_final <<<NBLK, 128, 0, stream>>>(xyz4, kidx, W1, W2, st1, gamma1, beta1,
                                     st2, gamma2, beta2, out);
}